// PeriodicDistance_67181878444399
// MI455X (gfx1250) — compile-verified
//
#include <hip/hip_runtime.h>
#include <math.h>

// ---------------------------------------------------------------------------
// PeriodicDistance neighbor build for MI455X (gfx1250, wave32).
//
// d2 tiles on the matrix pipe via V_WMMA_F32_16X16X4_F32:
//   d2[m][n] = x2[m] + x2[n] - 2*dot(p_m, p_n)
//   A[m][:]  = (-2x_m, -2y_m, -2z_m, 1)      (16x4, K=4)
//   B[:][n]  = ( x_n ,  y_n ,  z_n , x2_n)   (4x16)
//   C[m][n]  =  x2_m                          (preloaded accumulator)
// Validity folded in by poisoning the x2 term of invalid images (+1e30).
// B operand is software-pipelined (next tile's global_load_b64 issued before
// this tile's WMMA + top-32 scan, hiding the load latency). Per-lane top-32
// kept in registers with a branch-free cmp+cndmask chain that co-executes
// with the XDL WMMA pipe.
// ---------------------------------------------------------------------------

typedef __attribute__((ext_vector_type(2))) float v2f;
typedef __attribute__((ext_vector_type(8))) float v8f;

#define N_ATOMS   512
#define NIMG      27
#define M_PTS     (N_ATOMS * NIMG)      // 13824
#define N_TILES   (M_PTS / 16)          // 864
#define K_NBR     32
#define E_EDGES   (M_PTS * K_NBR)       // 442368
#define CUT       5.0f
#define CUT2      25.0f
#define POISON    1e30f
#define INF_F     __builtin_inff()

// ---- Phase 1: replicate points into 27 periodic images ---------------------
__global__ void __launch_bounds__(256)
build_images_kernel(const float* __restrict__ pos,
                    const float* __restrict__ box,
                    float4* __restrict__ cand,
                    float4* __restrict__ qvec) {
    int i = blockIdx.x * blockDim.x + threadIdx.x;
    if (i >= M_PTS) return;
    int a = i % N_ATOMS;            // original atom id (idmap)
    int s = i / N_ATOMS;            // shift index, a fastest (matches reference)

    const float offs[3] = {-1.0f, 1.0f, 0.0f};
    float bx = box[0], by = box[1], bz = box[2];
    float sx = offs[s % 3] * bx;
    float sy = offs[(s / 3) % 3] * by;
    float sz = offs[s / 9] * bz;

    float x = pos[a * 3 + 0] + sx;
    float y = pos[a * 3 + 1] + sy;
    float z = pos[a * 3 + 2] + sz;

    float ex = bx + CUT, ey = by + CUT, ez = bz + CUT;
    float u = x / ex, v = y / ey, w = z / ez;
    bool valid = (u * u + v * v + w * w) <= 1.0f;

    float x2 = x * x + y * y + z * z;
    float poison = valid ? 0.0f : POISON;

    cand[i] = make_float4(x, y, z, x2 + poison);
    qvec[i] = make_float4(-2.0f * x, -2.0f * y, -2.0f * z, 1.0f);
}

// ---- Phase 2: WMMA distance tiles + per-lane top-32 ------------------------
// One wave per block; lane L owns query q0+L. Each 16-candidate tile needs
// two WMMAs (query rows 0-15 and 16-31), transposed through 2KB of LDS.
__global__ void __launch_bounds__(32)
knn_wmma_kernel(const float4* __restrict__ candv,
                const float4* __restrict__ qvecv,
                const float* __restrict__ pos,
                float* __restrict__ out) {
    __shared__ float tile[32 * 16];        // d2 tile, [query_row][cand_col]
    __shared__ float pos_s[N_ATOMS * 3];   // original positions (epilogue)

    const float* cd = (const float*)candv;
    const float* qv = (const float*)qvecv;

    const int lane = threadIdx.x;       // 0..31
    const int q0 = blockIdx.x * 32;
    const int lo = lane & 15;
    const int kb = (lane < 16) ? 0 : 2; // A/B K-pair per lane half (ISA layout)

    // Stage pos[] into LDS (48 floats per lane).
#pragma unroll 1
    for (int i = lane; i < N_ATOMS * 3; i += 32)
        pos_s[i] = pos[i];

    // A operands: fixed for the whole block (reused every tile).
    v2f a0, a1;
    {
        int m = q0 + lo;
        a0.x = qv[m * 4 + kb];
        a0.y = qv[m * 4 + kb + 1];
        int m2 = m + 16;
        a1.x = qv[m2 * 4 + kb];
        a1.y = qv[m2 * 4 + kb + 1];
    }
    // C accumulator: x2[m] (+poison). C VGPR r: lanes0-15 -> M=r, hi -> M=r+8.
    v8f c0, c1;
    {
        int mrow = q0 + ((lane < 16) ? 0 : 8);
#pragma unroll
        for (int r = 0; r < 8; ++r) {
            c0[r] = cd[(mrow + r) * 4 + 3];
            c1[r] = cd[(mrow + 16 + r) * 4 + 3];
        }
    }

    // Per-lane sorted top-32 (ascending d2); strict '<' + ascending scan
    // order reproduces jax top_k's lowest-index tie-breaking.
    float hd[K_NBR];
    int hi[K_NBR];
#pragma unroll
    for (int k = 0; k < K_NBR; ++k) { hd[k] = INF_F; hi[k] = 0; }

    // Prime the B-operand pipeline with tile 0.
    v2f bcur;
    bcur.x = cd[lo * 4 + kb];
    bcur.y = cd[lo * 4 + kb + 1];

    __syncthreads();   // pos_s ready

    for (int t = 0; t < N_TILES; ++t) {
        const int n0 = t * 16;

        // Issue next tile's B load now; wait lands before NEXT iteration's
        // WMMA, hidden behind this tile's insertion scan.
        v2f bnext = bcur;
        if (t + 1 < N_TILES) {
            int n = n0 + 16 + lo;
            bnext.x = cd[n * 4 + kb];
            bnext.y = cd[n * 4 + kb + 1];
            __builtin_prefetch(&cd[(n + 16) * 4], 0, 3);  // global_prefetch_b8
        }

        // D = A x B + C  -> 16x16 d2 tile (two query row-tiles)
        v8f d0 = __builtin_amdgcn_wmma_f32_16x16x4_f32(
            false, a0, false, bcur, (short)0, c0, false, false);
        v8f d1 = __builtin_amdgcn_wmma_f32_16x16x4_f32(
            false, a1, false, bcur, (short)0, c1, false, false);

        // Transpose through LDS so lane L sees its query row contiguously.
        int half8 = (lane < 16) ? 0 : 8;
#pragma unroll
        for (int r = 0; r < 8; ++r) {
            tile[(half8 + r) * 16 + lo] = d0[r];
            tile[(16 + half8 + r) * 16 + lo] = d1[r];
        }
        __syncthreads();

        // Scan this lane's query row: 4x ds_load_b128, guarded insertion.
#pragma unroll 1
        for (int jj = 0; jj < 16; jj += 4) {
            float4 dv = *(const float4*)&tile[lane * 16 + jj];
#pragma unroll
            for (int u = 0; u < 4; ++u) {
                float d2 = (u == 0) ? dv.x : (u == 1) ? dv.y
                         : (u == 2) ? dv.z : dv.w;
                d2 = fmaxf(d2, 0.0f);
                float v = (d2 <= CUT2) ? d2 : INF_F;
                // Divergence-friendly skip: almost all candidates fail here.
                if (v < hd[K_NBR - 1]) {
                    float cv = v;
                    int ci = n0 + jj + u;
                    // Branch-free sorted insert; values are never NaN, so
                    // plain selects (1 v_cmp + 4 v_cndmask) beat min/max.
#pragma unroll
                    for (int k = 0; k < K_NBR; ++k) {
                        bool sw = cv < hd[k];
                        float mn = sw ? cv    : hd[k];
                        float mx = sw ? hd[k] : cv;
                        int ni   = sw ? ci    : hi[k];
                        int xi   = sw ? hi[k] : ci;
                        hd[k] = mn; cv = mx;
                        hi[k] = ni; ci = xi;
                    }
                }
            }
        }
        __syncthreads();

        bcur = bnext;
    }

    // ---- Emit edges: [row|col](2E), weight(E), edge_vec(E*3), nv(E) -------
    const int E = E_EDGES;
    int m = q0 + lane;
    int colAtom = m & (N_ATOMS - 1);    // idmap[query image] = m % N
#pragma unroll 1
    for (int k = 0; k < K_NBR; ++k) {
        int e = m * K_NBR + k;
        bool nv = hd[k] < POISON;        // finite (<=25) iff valid edge
        int nbr = hi[k];
        int row = nv ? (nbr & (N_ATOMS - 1)) : 0;
        int col = nv ? colAtom : 0;

        float ex = 0.0f, ey = 0.0f, ez = 0.0f;
        if (nv) {
            ex = pos_s[row * 3 + 0] - pos_s[col * 3 + 0];
            ey = pos_s[row * 3 + 1] - pos_s[col * 3 + 1];
            ez = pos_s[row * 3 + 2] - pos_s[col * 3 + 2];
        }
        bool live = nv && (row != col);  // self-loops carry zero weight
        float wgt = live ? sqrtf(ex * ex + ey * ey + ez * ez) : 0.0f;

        out[e]                 = (float)row;     // edge_index[0]
        out[E + e]             = (float)col;     // edge_index[1]
        out[2 * E + e]         = wgt;            // edge_weight
        out[3 * E + e * 3 + 0] = ex;             // edge_vec
        out[3 * E + e * 3 + 1] = ey;
        out[3 * E + e * 3 + 2] = ez;
        out[6 * E + e]         = nv ? 1.0f : 0.0f;  // nv mask
    }
}

// ---------------------------------------------------------------------------
extern "C" void kernel_launch(void* const* d_in, const int* in_sizes, int n_in,
                              void* d_out, int out_size, void* d_ws, size_t ws_size,
                              hipStream_t stream) {
    const float* pos = (const float*)d_in[0];   // [512,3] f32
    // d_in[1]: batch (all zeros, single system) -- unused
    const float* box = (const float*)d_in[2];   // [3] f32

    float4* cand = (float4*)d_ws;               // M_PTS float4
    float4* qvec = cand + M_PTS;                // M_PTS float4  (~432 KB total)

    build_images_kernel<<<(M_PTS + 255) / 256, 256, 0, stream>>>(
        pos, box, cand, qvec);

    knn_wmma_kernel<<<M_PTS / 32, 32, 0, stream>>>(
        cand, qvec, pos, (float*)d_out);
}